// GCM_19250043421437
// MI455X (gfx1250) — compile-verified
//
#include <hip/hip_runtime.h>

#define BATCH 512
#define TLEN  512
#define NVAR  64
#define RANK  16

typedef __attribute__((ext_vector_type(2))) float v2f;
typedef __attribute__((ext_vector_type(4))) float v4f;
typedef __attribute__((ext_vector_type(8))) float v8f;

// Prefer the gfx1250 hardware V_TANH_F32 transcendental if the toolchain
// exposes it; fall back to libm tanhf otherwise.
#if __has_builtin(__builtin_amdgcn_tanhf)
  #define TANHF(x) __builtin_amdgcn_tanhf(x)
#elif __has_builtin(__builtin_amdgcn_tanh_f32)
  #define TANHF(x) __builtin_amdgcn_tanh_f32(x)
#else
  #define TANHF(x) tanhf(x)
#endif

// Bs LDS layout [k16][v64][o16], XOR-swizzled so both the v-strided staging
// stores and the o-strided WMMA-operand reads are bank-conflict-free.
#define BS_IDX(k, v, o)   ((((k) * 64 + (v)) * 16) + (((o) ^ (((v) >> 2) & 15))))
// outs LDS layout [b16][o16][v64], XOR-swizzled: WMMA D writes vary o with v
// fixed; the coalesced write-out pass varies v with o fixed.
#define OUTS_IDX(b, o, v) ((((b) * 16 + (o)) * 64) + (((v) ^ ((((o) & 15) << 2) & 63))))

// ---------------------------------------------------------------------------
// Stage 1: tmp[v][b][k] = sum_i tanh(g_v * x[b][i][v]) * lora_A[i][k]
// grid = (BATCH/16 b-tiles, NVAR/16 v-groups); block = 256 (8 waves, 2 v/wave)
// ---------------------------------------------------------------------------
__global__ __launch_bounds__(256) void gcm_stage1(
    const float* __restrict__ x, const float* __restrict__ gating,
    const float* __restrict__ lora_A, float* __restrict__ tmp)
{
  __shared__ float xs[32 * 16 * 17];  // [i32][v16][b16 pad->17]
  __shared__ float as[32 * 16];       // [i32][k16]

  const int tid  = threadIdx.x;
  const int lane = tid & 31;
  const int wave = tid >> 5;
  const int b0   = blockIdx.x * 16;
  const int v0   = blockIdx.y * 16;

  const int m    = lane & 15;             // M (row) / N (col) index for this lane
  const int koff = (lane < 16) ? 0 : 2;   // K sub-offset per lane half
  const int vla  = wave * 2;              // two local v's per wave
  const int vlb  = vla + 1;
  const float ga = gating[v0 + vla];
  const float gb = gating[v0 + vlb];

  // Per-thread fixed (b, i, v) decomposition for the 8 staging loads per stage.
  const int sv = (tid * 4) & 15;          // multiple of 4
  const int si = (tid >> 2) & 31;         // i within 32-chunk (tid*4 >> 4)
  // b advances by (tid + r*256)*4 >> 9 = 2*r + (tid >> 7)
  const int sb = tid >> 7;

  v8f acc0 = {0.f, 0.f, 0.f, 0.f, 0.f, 0.f, 0.f, 0.f};
  v8f acc1 = {0.f, 0.f, 0.f, 0.f, 0.f, 0.f, 0.f, 0.f};

  for (int s = 0; s < TLEN / 32; ++s) {
    const int i0 = s * 32;
    __syncthreads();  // previous stage fully consumed before overwrite

    // Stage x[b0..b0+15][i0..i0+31][v0..v0+15] -> xs[i][v][b].
    // Fixed trip count + unroll => 8 independent b128 loads in flight.
    v4f vals[8];
    #pragma unroll
    for (int r = 0; r < 8; ++r) {
      const int b = sb + 2 * r;
      vals[r] = *reinterpret_cast<const v4f*>(
          &x[((size_t)(b0 + b) * TLEN + (i0 + si)) * NVAR + v0 + sv]);
    }
    #pragma unroll
    for (int r = 0; r < 8; ++r) {
      const int b = sb + 2 * r;
      xs[((si * 16) + sv + 0) * 17 + b] = vals[r].x;
      xs[((si * 16) + sv + 1) * 17 + b] = vals[r].y;
      xs[((si * 16) + sv + 2) * 17 + b] = vals[r].z;
      xs[((si * 16) + sv + 3) * 17 + b] = vals[r].w;
    }
    // Stage lora_A rows i0..i0+31 (shared by all 8 waves).
    #pragma unroll
    for (int r = 0; r < 2; ++r) {
      const int e = tid + r * 256;
      const int k = e & 15;
      const int i = e >> 4;
      as[i * 16 + k] = lora_A[(size_t)(i0 + i) * RANK + k];
    }
    __syncthreads();

    #pragma unroll
    for (int ic = 0; ic < 8; ++ic) {
      const int il = ic * 4 + koff;   // local K-pair base row for this lane
      v2f bmat;                       // B operand: lora_A[il..il+1][m]
      bmat.x = as[(il + 0) * 16 + m];
      bmat.y = as[(il + 1) * 16 + m];
      v2f a0, a1;                     // A operand: tanh(g * x[b=m][il..il+1][v])
      a0.x = TANHF(ga * xs[((il + 0) * 16 + vla) * 17 + m]);
      a0.y = TANHF(ga * xs[((il + 1) * 16 + vla) * 17 + m]);
      a1.x = TANHF(gb * xs[((il + 0) * 16 + vlb) * 17 + m]);
      a1.y = TANHF(gb * xs[((il + 1) * 16 + vlb) * 17 + m]);
      acc0 = __builtin_amdgcn_wmma_f32_16x16x4_f32(false, a0, false, bmat,
                                                   (short)0, acc0, false, false);
      acc1 = __builtin_amdgcn_wmma_f32_16x16x4_f32(false, a1, false, bmat,
                                                   (short)0, acc1, false, false);
    }
  }

  // D layout: VGPR r holds M=r (lanes 0-15) / M=8+r (lanes 16-31), N=lane%16.
  const int mrow = (lane >> 4) * 8;
  const int n    = lane & 15;
  #pragma unroll
  for (int r = 0; r < 8; ++r) {
    tmp[((size_t)(v0 + vla) * BATCH + b0 + mrow + r) * RANK + n] = acc0[r];
    tmp[((size_t)(v0 + vlb) * BATCH + b0 + mrow + r) * RANK + n] = acc1[r];
  }
}

// ---------------------------------------------------------------------------
// Stage 2: out[b][o][v] = x[b][o][v] + bias[o][v] + sum_k tmp[v][b][k]*B[k][o][v]
// grid = (BATCH/16 b-tiles, TLEN/16 o-tiles); block = 1024 (32 waves, 2 v/wave)
// ---------------------------------------------------------------------------
__global__ __launch_bounds__(1024) void gcm_stage2(
    const float* __restrict__ x, const float* __restrict__ bias,
    const float* __restrict__ lora_B, const float* __restrict__ tmp,
    float* __restrict__ out)
{
  __shared__ float smem[16 * 64 * 16];  // 64 KB: Bs tile, reused as outs tile
  float* Bs   = smem;
  float* outs = smem;

  const int tid  = threadIdx.x;
  const int lane = tid & 31;
  const int wave = tid >> 5;
  const int b0   = blockIdx.x * 16;
  const int o0   = blockIdx.y * 16;
  const int m    = lane & 15;
  const int koff = (lane < 16) ? 0 : 2;
  const int va   = wave * 2;            // two (global) v's per wave
  const int vb   = va + 1;

  // Per-thread fixed decomposition for the 4 staging iterations.
  const int sv = (tid * 4) & 63;        // multiple of 4
  const int so = (tid >> 4) & 15;       // (tid*4 >> 6) & 15
  // outer index advances by (tid + r*1024)*4 >> 10 = 4*r + (tid >> 8)
  const int sk = tid >> 8;

  // Stage lora_B[k][o0+so][sv..sv+3] -> Bs, 4 independent b128 loads.
  {
    v4f vals[4];
    #pragma unroll
    for (int r = 0; r < 4; ++r) {
      const int k = sk + 4 * r;
      vals[r] = *reinterpret_cast<const v4f*>(
          &lora_B[((size_t)k * TLEN + o0 + so) * NVAR + sv]);
    }
    #pragma unroll
    for (int r = 0; r < 4; ++r) {
      const int k = sk + 4 * r;
      Bs[BS_IDX(k, sv + 0, so)] = vals[r].x;
      Bs[BS_IDX(k, sv + 1, so)] = vals[r].y;
      Bs[BS_IDX(k, sv + 2, so)] = vals[r].z;
      Bs[BS_IDX(k, sv + 3, so)] = vals[r].w;
    }
  }
  __syncthreads();

  v8f acc0 = {0.f, 0.f, 0.f, 0.f, 0.f, 0.f, 0.f, 0.f};
  v8f acc1 = {0.f, 0.f, 0.f, 0.f, 0.f, 0.f, 0.f, 0.f};
  #pragma unroll
  for (int kc = 0; kc < 4; ++kc) {
    const int k = kc * 4 + koff;
    v2f a0, a1, bm0, bm1;
    // A operand: TMP_v[b = m][k..k+1], direct from L2-hot workspace.
    a0.x = tmp[((size_t)va * BATCH + b0 + m) * RANK + k + 0];
    a0.y = tmp[((size_t)va * BATCH + b0 + m) * RANK + k + 1];
    a1.x = tmp[((size_t)vb * BATCH + b0 + m) * RANK + k + 0];
    a1.y = tmp[((size_t)vb * BATCH + b0 + m) * RANK + k + 1];
    // B operand: B_v[k..k+1][o = m] from LDS.
    bm0.x = Bs[BS_IDX(k + 0, va, m)];
    bm0.y = Bs[BS_IDX(k + 1, va, m)];
    bm1.x = Bs[BS_IDX(k + 0, vb, m)];
    bm1.y = Bs[BS_IDX(k + 1, vb, m)];
    acc0 = __builtin_amdgcn_wmma_f32_16x16x4_f32(false, a0, false, bm0,
                                                 (short)0, acc0, false, false);
    acc1 = __builtin_amdgcn_wmma_f32_16x16x4_f32(false, a1, false, bm1,
                                                 (short)0, acc1, false, false);
  }
  __syncthreads();  // all Bs reads complete before the buffer is reused

  // Park WMMA results in LDS (per-wave v's are disjoint slices).
  const int mrow = (lane >> 4) * 8;
  const int n    = lane & 15;
  #pragma unroll
  for (int r = 0; r < 8; ++r) {
    outs[OUTS_IDX(mrow + r, n, va)] = acc0[r];
    outs[OUTS_IDX(mrow + r, n, vb)] = acc1[r];
  }
  __syncthreads();

  // Residual + bias + store: fully coalesced v4f along v; non-temporal
  // stores keep the 64 MB output stream from evicting x (re-read from L2).
  #pragma unroll
  for (int r = 0; r < 4; ++r) {
    const int b = sk + 4 * r;   // (tid + r*1024)*4 >> 10
    const size_t g = ((size_t)(b0 + b) * TLEN + (o0 + so)) * NVAR + sv;
    const v4f xv = *reinterpret_cast<const v4f*>(&x[g]);
    const v4f bv = *reinterpret_cast<const v4f*>(
        &bias[(size_t)(o0 + so) * NVAR + sv]);
    v4f ov;
    ov.x = xv.x + bv.x + outs[OUTS_IDX(b, so, sv + 0)];
    ov.y = xv.y + bv.y + outs[OUTS_IDX(b, so, sv + 1)];
    ov.z = xv.z + bv.z + outs[OUTS_IDX(b, so, sv + 2)];
    ov.w = xv.w + bv.w + outs[OUTS_IDX(b, so, sv + 3)];
    __builtin_nontemporal_store(ov, reinterpret_cast<v4f*>(&out[g]));
  }
}

extern "C" void kernel_launch(void* const* d_in, const int* in_sizes, int n_in,
                              void* d_out, int out_size, void* d_ws, size_t ws_size,
                              hipStream_t stream) {
  (void)in_sizes; (void)n_in; (void)out_size; (void)ws_size;
  const float* x      = (const float*)d_in[0];
  const float* gating = (const float*)d_in[1];
  const float* bias   = (const float*)d_in[2];
  const float* lora_A = (const float*)d_in[3];
  const float* lora_B = (const float*)d_in[4];
  float* out = (float*)d_out;
  float* tmp = (float*)d_ws;  // NVAR*BATCH*RANK floats = 2 MiB scratch

  dim3 g1(BATCH / 16, NVAR / 16);
  gcm_stage1<<<g1, dim3(256), 0, stream>>>(x, gating, lora_A, tmp);

  dim3 g2(BATCH / 16, TLEN / 16);
  gcm_stage2<<<g2, dim3(1024), 0, stream>>>(x, bias, lora_B, tmp, out);
}